// Frontend_51049981281419
// MI455X (gfx1250) — compile-verified
//
#include <hip/hip_runtime.h>
#include <math.h>

#define Bb 8
#define Tt 512
#define Cc 4
#define Ff 257
#define Aa 320
#define BF (Bb*Ff)   /* 2056 */
#define KC 24
#define TP 504       /* T - bdelay - btaps + 1 */

typedef float v2f __attribute__((ext_vector_type(2)));
typedef float v8f __attribute__((ext_vector_type(8)));

static __device__ inline v8f wmma_f32(v2f a, v2f b, v8f c) {
  // D(16x16,f32) += A(16x4,f32) * B(4x16,f32)
  return __builtin_amdgcn_wmma_f32_16x16x4_f32(false, a, false, b, (short)0, c, false, false);
}

// ---------------------------------------------------------------------------
// K1: repack x (B,T,C,F) -> planar Y (B,F,C,T) real/imag
// ---------------------------------------------------------------------------
__global__ void pack_kernel(const float* __restrict__ xr, const float* __restrict__ xi,
                            float* __restrict__ Yr, float* __restrict__ Yi) {
  int gid = blockIdx.x * 256 + threadIdx.x;
  const int N = Bb * Tt * Cc * Ff;
  if (gid >= N) return;
  int f = gid % Ff; int r = gid / Ff;
  int c = r % Cc;   r /= Cc;
  int t = r % Tt;   int b = r / Tt;
  size_t dst = (((size_t)b * Ff + f) * Cc + c) * Tt + t;
  Yr[dst] = xr[gid];
  Yi[dst] = xi[gid];
}

// ---------------------------------------------------------------------------
// K2: inv_power = 1/clip(mean_c(|Y|^2 * mask), 1e-7); mbar = mean_c(mask)
// ---------------------------------------------------------------------------
__global__ void invp_kernel(const float* __restrict__ Yr, const float* __restrict__ Yi,
                            const float* __restrict__ mask,
                            float* __restrict__ invp, float* __restrict__ mbar) {
  int gid = blockIdx.x * 256 + threadIdx.x;
  const int N = BF * Tt;
  if (gid >= N) return;
  int t = gid % Tt; int bf = gid / Tt;
  size_t base = (size_t)bf * Cc * Tt + t;
  float p = 0.f, mm = 0.f;
#pragma unroll
  for (int c = 0; c < Cc; ++c) {
    float yr = Yr[base + c * Tt], yi = Yi[base + c * Tt];
    float m = mask[base + c * Tt];
    p += (yr * yr + yi * yi) * m;
    mm += m;
  }
  p *= 0.25f; mm *= 0.25f;
  invp[gid] = 1.f / fmaxf(p, 1e-7f);
  mbar[gid] = mm;
}

// ---------------------------------------------------------------------------
// K3: weighted complex Gram (covariance) per (b,f) via V_WMMA_F32_16X16X4_F32
// cov[kd,le] = sum_t Z[kd,t] * conj(Z[le,t]) * w[t],  Z[k*4+d,t]=Y[d,t+off[k]]
// off = {8,4,3,2,1,0}; w[t] = inv_power[t+8]
// ---------------------------------------------------------------------------
static __device__ inline void store_tile(float* __restrict__ dst, const v8f a,
                                         int rt, int ct, int half, int lm) {
#pragma unroll
  for (int v = 0; v < 8; ++v) {
    int row = rt * 16 + v + 8 * half;
    int col = ct * 16 + lm;
    if (row < KC && col < KC) dst[row * KC + col] = a[v];
  }
}

__global__ void cov_kernel(const float* __restrict__ Yr, const float* __restrict__ Yi,
                           const float* __restrict__ invp,
                           float* __restrict__ covr, float* __restrict__ covi) {
  __shared__ float sYr[Cc * Tt];
  __shared__ float sYi[Cc * Tt];
  __shared__ float sW[Tt];
  const int bf = blockIdx.x;
  const int lane = threadIdx.x;
  const float* gYr = Yr + (size_t)bf * Cc * Tt;
  const float* gYi = Yi + (size_t)bf * Cc * Tt;
  const float* gW  = invp + (size_t)bf * Tt;
  for (int i = lane; i < Cc * Tt; i += 32) { sYr[i] = gYr[i]; sYi[i] = gYi[i]; }
  for (int i = lane; i < Tt; i += 32) sW[i] = gW[i];
  __syncthreads();

  const int half = lane >> 4, lm = lane & 15;
  // row/col index -> (channel d, time offset off[k]); cols share the mapping.
  const int k0 = lm >> 2, d0 = lm & 3;
  const int o0 = (k0 == 0) ? 8 : (5 - k0);
  const int m1 = lm + 16;
  const float v1 = (m1 < KC) ? 1.f : 0.f;
  const int m1c = (m1 < KC) ? m1 : (KC - 1);
  const int k1 = m1c >> 2, d1 = m1c & 3;
  const int o1 = (k1 == 0) ? 8 : (5 - k1);

  v8f cr00 = {}, cr01 = {}, cr10 = {}, cr11 = {};
  v8f ci00 = {}, ci01 = {}, ci10 = {}, ci11 = {};

  for (int t0 = 0; t0 < TP; t0 += 4) {
    const int ta = t0 + 2 * half;           // A/B K-split across lane halves
    const int i0 = d0 * Tt + ta + o0;
    const int i1 = d1 * Tt + ta + o1;
    v2f a0r = { sYr[i0], sYr[i0 + 1] };
    v2f a0i = { sYi[i0], sYi[i0 + 1] };
    v2f a1r = { v1 * sYr[i1], v1 * sYr[i1 + 1] };
    v2f a1i = { v1 * sYi[i1], v1 * sYi[i1 + 1] };
    const float w0 = sW[ta + 8], w1 = sW[ta + 9];
    v2f b0r = { w0 * a0r.x, w1 * a0r.y };   // B[t,n] = w_t * Z[n,t]
    v2f b0i = { w0 * a0i.x, w1 * a0i.y };
    v2f b1r = { w0 * a1r.x, w1 * a1r.y };
    v2f b1i = { w0 * a1i.x, w1 * a1i.y };
    v2f n0i = -b0i;
    v2f n1i = -b1i;

    cr00 = wmma_f32(a0r, b0r, cr00); cr00 = wmma_f32(a0i, b0i, cr00);
    ci00 = wmma_f32(a0i, b0r, ci00); ci00 = wmma_f32(a0r, n0i, ci00);
    cr01 = wmma_f32(a0r, b1r, cr01); cr01 = wmma_f32(a0i, b1i, cr01);
    ci01 = wmma_f32(a0i, b1r, ci01); ci01 = wmma_f32(a0r, n1i, ci01);
    cr10 = wmma_f32(a1r, b0r, cr10); cr10 = wmma_f32(a1i, b0i, cr10);
    ci10 = wmma_f32(a1i, b0r, ci10); ci10 = wmma_f32(a1r, n0i, ci10);
    cr11 = wmma_f32(a1r, b1r, cr11); cr11 = wmma_f32(a1i, b1i, cr11);
    ci11 = wmma_f32(a1i, b1r, ci11); ci11 = wmma_f32(a1r, n1i, ci11);
  }

  float* dr = covr + (size_t)bf * (KC * KC);
  float* di = covi + (size_t)bf * (KC * KC);
  store_tile(dr, cr00, 0, 0, half, lm); store_tile(dr, cr01, 0, 1, half, lm);
  store_tile(dr, cr10, 1, 0, half, lm); store_tile(dr, cr11, 1, 1, half, lm);
  store_tile(di, ci00, 0, 0, half, lm); store_tile(di, ci01, 0, 1, half, lm);
  store_tile(di, ci10, 1, 0, half, lm); store_tile(di, ci11, 1, 1, half, lm);
}

// ---------------------------------------------------------------------------
// K4: psd[b,f,c,e] = sum_t Y[c,t]*conj(Y[e,t])*mbar[t]   (4x4 complex)
// ---------------------------------------------------------------------------
__global__ void psd_kernel(const float* __restrict__ Yr, const float* __restrict__ Yi,
                           const float* __restrict__ mbar,
                           float* __restrict__ psdr, float* __restrict__ psdi) {
  int gid = blockIdx.x * 256 + threadIdx.x;
  const int N = BF * 16;
  if (gid >= N) return;
  int bf = gid >> 4; int ce = gid & 15; int c = ce >> 2; int e = ce & 3;
  const float* ycr = Yr + ((size_t)bf * Cc + c) * Tt;
  const float* yci = Yi + ((size_t)bf * Cc + c) * Tt;
  const float* yer = Yr + ((size_t)bf * Cc + e) * Tt;
  const float* yei = Yi + ((size_t)bf * Cc + e) * Tt;
  const float* mb = mbar + (size_t)bf * Tt;
  float sr = 0.f, si = 0.f;
  for (int t = 0; t < Tt; ++t) {
    float ar = ycr[t], ai = yci[t], br = yer[t], bi = yei[t], m = mb[t];
    sr += m * (ar * br + ai * bi);
    si += m * (ai * br - ar * bi);
  }
  psdr[gid] = sr;
  psdi[gid] = si;
}

// ---------------------------------------------------------------------------
// K5a: attention feature + MLP -> e[b,c]
// ---------------------------------------------------------------------------
__global__ void attn_kernel(const float* __restrict__ psdr, const float* __restrict__ psdi,
                            const float* __restrict__ Wp, const float* __restrict__ bp,
                            const float* __restrict__ wg, const float* __restrict__ bg,
                            float* __restrict__ eout) {
  __shared__ float feat[Ff];
  __shared__ float red[256];
  int bc = blockIdx.x; int b = bc >> 2; int c = bc & 3; int tid = threadIdx.x;
  for (int f = tid; f < Ff; f += 256) {
    size_t base = ((size_t)(b * Ff + f)) * 16 + c * 4;
    float sr = 0.f, si = 0.f;
#pragma unroll
    for (int e = 0; e < 4; ++e) {
      if (e != c) { sr += psdr[base + e]; si += psdi[base + e]; }
    }
    sr *= (1.f / 3.f); si *= (1.f / 3.f);
    feat[f] = sqrtf(sr * sr + si * si);
  }
  __syncthreads();
  float acc = 0.f;
  for (int a = tid; a < Aa; a += 256) {
    float s = bp[a];
    for (int f = 0; f < Ff; ++f) s += feat[f] * Wp[f * Aa + a];
    acc += tanhf(s) * wg[a];
  }
  red[tid] = acc;
  __syncthreads();
  for (int st = 128; st > 0; st >>= 1) {
    if (tid < st) red[tid] += red[tid + st];
    __syncthreads();
  }
  if (tid == 0) eout[bc] = red[0] + bg[0];
}

// K5b: u = softmax(2*e) over C
__global__ void softmax_kernel(const float* __restrict__ eb, float* __restrict__ ub) {
  int b = threadIdx.x;
  if (b < Bb) {
    float m = -1e30f;
    for (int c = 0; c < 4; ++c) m = fmaxf(m, 2.f * eb[b * 4 + c]);
    float ex[4]; float s = 0.f;
    for (int c = 0; c < 4; ++c) { ex[c] = expf(2.f * eb[b * 4 + c] - m); s += ex[c]; }
    for (int c = 0; c < 4; ++c) ub[b * 4 + c] = ex[c] / s;
  }
}

// ---------------------------------------------------------------------------
// K6: per (b,f): solve Rf X = I[:, :4] (complex Gauss-Jordan, partial pivot),
// num = X @ Phi, ws = num/(trace+eps), filt = conj(ws @ u)
// ---------------------------------------------------------------------------
__global__ void solve_kernel(const float* __restrict__ covr, const float* __restrict__ covi,
                             const float* __restrict__ psdr, const float* __restrict__ psdi,
                             const float* __restrict__ ubuf,
                             float* __restrict__ filtr, float* __restrict__ filti) {
  __shared__ float Mr[KC][28], Mi[KC][28];
  __shared__ float facr[KC], faci[KC];
  __shared__ float nr[KC][4], ni[KC][4];
  __shared__ int piv;
  __shared__ float trv[2];
  const int bf = blockIdx.x;
  const int lane = threadIdx.x;
  const int b = bf / Ff;
  const size_t cb = (size_t)bf * (KC * KC);

  if (lane < KC) {
    for (int i = 0; i < KC; ++i) { Mr[i][lane] = covr[cb + i * KC + lane]; Mi[i][lane] = covi[cb + i * KC + lane]; }
  } else if (lane < 28) {
    int q = lane - 24;
    for (int i = 0; i < KC; ++i) { Mr[i][lane] = (i == q) ? 1.f : 0.f; Mi[i][lane] = 0.f; }
  }
  __syncthreads();

  for (int p = 0; p < KC; ++p) {
    if (lane == 0) {
      int best = p; float bm = -1.f;
      for (int i = p; i < KC; ++i) {
        float m2 = Mr[i][p] * Mr[i][p] + Mi[i][p] * Mi[i][p];
        if (m2 > bm) { bm = m2; best = i; }
      }
      piv = best;
    }
    __syncthreads();
    int r = piv;
    if (lane < 28 && r != p) {
      float a = Mr[p][lane], bb = Mi[p][lane];
      Mr[p][lane] = Mr[r][lane]; Mi[p][lane] = Mi[r][lane];
      Mr[r][lane] = a; Mi[r][lane] = bb;
    }
    __syncthreads();
    float pr = Mr[p][p], pii = Mi[p][p];
    __syncthreads();
    float den = pr * pr + pii * pii;
    float idr = pr / den, idi = -pii / den;
    if (lane < 28) {
      float a = Mr[p][lane], bb = Mi[p][lane];
      Mr[p][lane] = a * idr - bb * idi;
      Mi[p][lane] = a * idi + bb * idr;
    }
    __syncthreads();
    if (lane < KC) { facr[lane] = Mr[lane][p]; faci[lane] = Mi[lane][p]; }
    __syncthreads();
    if (lane < 28) {
      float mpr = Mr[p][lane], mpi = Mi[p][lane];
      for (int i = 0; i < KC; ++i) {
        if (i == p) continue;
        float fr = facr[i], fi = faci[i];
        Mr[i][lane] -= fr * mpr - fi * mpi;
        Mi[i][lane] -= fr * mpi + fi * mpr;
      }
    }
    __syncthreads();
  }

  const size_t pb = (size_t)bf * 16;
  if (lane < KC) {
    for (int cc = 0; cc < 4; ++cc) {
      float ar = 0.f, ai = 0.f;
      for (int q = 0; q < 4; ++q) {
        float xr = Mr[lane][24 + q], xi = Mi[lane][24 + q];
        float phr = psdr[pb + q * 4 + cc], phi = psdi[pb + q * 4 + cc];
        ar += xr * phr - xi * phi;
        ai += xr * phi + xi * phr;
      }
      nr[lane][cc] = ar; ni[lane][cc] = ai;
    }
  }
  __syncthreads();
  if (lane == 0) {
    float tr = 1e-15f, ti = 0.f;
    for (int cc = 0; cc < 4; ++cc) { tr += nr[cc][cc]; ti += ni[cc][cc]; }
    trv[0] = tr; trv[1] = ti;
  }
  __syncthreads();
  if (lane < KC) {
    float tr = trv[0], ti = trv[1];
    float dd = tr * tr + ti * ti;
    float br = 0.f, bi = 0.f;
    for (int cc = 0; cc < 4; ++cc) {
      float ar = nr[lane][cc], ai = ni[lane][cc];
      float wr = (ar * tr + ai * ti) / dd;
      float wi = (ai * tr - ar * ti) / dd;
      float uu = ubuf[b * 4 + cc];
      br += wr * uu; bi += wi * uu;
    }
    filtr[(size_t)bf * KC + lane] = br;   // conj
    filti[(size_t)bf * KC + lane] = -bi;
  }
}

// ---------------------------------------------------------------------------
// K7: enh[b,f,t] = sum_{k,c} Y[c, t+off[k]-8] * filt_conj[k*4+c]; write (4,B,T,F)
// ---------------------------------------------------------------------------
__global__ void filt_kernel(const float* __restrict__ Yr, const float* __restrict__ Yi,
                            const float* __restrict__ filtr, const float* __restrict__ filti,
                            float* __restrict__ out, int plane0) {
  int gid = blockIdx.x * 256 + threadIdx.x;
  const int N = BF * Tt;
  if (gid >= N) return;
  int t = gid & (Tt - 1); int bf = gid >> 9;
  int b = bf / Ff, f = bf % Ff;
  const float* yr = Yr + (size_t)bf * Cc * Tt;
  const float* yi = Yi + (size_t)bf * Cc * Tt;
  const float* fr = filtr + (size_t)bf * KC;
  const float* fi = filti + (size_t)bf * KC;
  const int offm8[6] = {0, -4, -5, -6, -7, -8};
  float er = 0.f, ei = 0.f;
#pragma unroll
  for (int k = 0; k < 6; ++k) {
    int tt = t + offm8[k];
    if (tt >= 0) {
#pragma unroll
      for (int c = 0; c < 4; ++c) {
        float a = yr[c * Tt + tt], bb = yi[c * Tt + tt];
        float cr = fr[k * 4 + c], ci = fi[k * 4 + c];
        er += a * cr - bb * ci;
        ei += a * ci + bb * cr;
      }
    }
  }
  out[(((size_t)plane0 * Bb + b) * Tt + t) * Ff + f] = er;
  out[(((size_t)(plane0 + 1) * Bb + b) * Tt + t) * Ff + f] = ei;
}

// ---------------------------------------------------------------------------
extern "C" void kernel_launch(void* const* d_in, const int* in_sizes, int n_in,
                              void* d_out, int out_size, void* d_ws, size_t ws_size,
                              hipStream_t stream) {
  (void)in_sizes; (void)n_in; (void)out_size; (void)ws_size;
  const float* xr    = (const float*)d_in[0];
  const float* xi    = (const float*)d_in[1];
  const float* mask1 = (const float*)d_in[2];
  const float* mask2 = (const float*)d_in[3];
  const float* Wp    = (const float*)d_in[4];
  const float* bp    = (const float*)d_in[5];
  const float* wg    = (const float*)d_in[6];
  const float* bg    = (const float*)d_in[7];
  float* out = (float*)d_out;

  float* W = (float*)d_ws;
  size_t o = 0;
  const size_t NY   = (size_t)Bb * Ff * Cc * Tt;
  const size_t NBFT = (size_t)Bb * Ff * Tt;
  float* Yr    = W + o; o += NY;
  float* Yi    = W + o; o += NY;
  float* invp  = W + o; o += NBFT;
  float* mbar  = W + o; o += NBFT;
  float* covr  = W + o; o += (size_t)BF * KC * KC;
  float* covi  = W + o; o += (size_t)BF * KC * KC;
  float* psdr  = W + o; o += (size_t)BF * 16;
  float* psdi  = W + o; o += (size_t)BF * 16;
  float* ebuf  = W + o; o += 32;
  float* ubuf  = W + o; o += 32;
  float* filtr = W + o; o += (size_t)BF * KC;
  float* filti = W + o; o += (size_t)BF * KC;

  const int n1 = Bb * Tt * Cc * Ff;
  pack_kernel<<<(n1 + 255) / 256, 256, 0, stream>>>(xr, xi, Yr, Yi);

  for (int br = 0; br < 2; ++br) {
    const float* mask = br ? mask2 : mask1;
    const int n2 = BF * Tt;
    invp_kernel<<<(n2 + 255) / 256, 256, 0, stream>>>(Yr, Yi, mask, invp, mbar);
    cov_kernel<<<BF, 32, 0, stream>>>(Yr, Yi, invp, covr, covi);
    const int n3 = BF * 16;
    psd_kernel<<<(n3 + 255) / 256, 256, 0, stream>>>(Yr, Yi, mbar, psdr, psdi);
    attn_kernel<<<Bb * Cc, 256, 0, stream>>>(psdr, psdi, Wp, bp, wg, bg, ebuf);
    softmax_kernel<<<1, 32, 0, stream>>>(ebuf, ubuf);
    solve_kernel<<<BF, 32, 0, stream>>>(covr, covi, psdr, psdi, ubuf, filtr, filti);
    filt_kernel<<<(n2 + 255) / 256, 256, 0, stream>>>(Yr, Yi, filtr, filti, out, 2 * br);
  }
}